// DMTetGeometry_76424648065138
// MI455X (gfx1250) — compile-verified
//
#include <hip/hip_runtime.h>
#include <hip/hip_bf16.h>

// ---------------------------------------------------------------------------
// DMTet (marching tetrahedra) for gfx1250 / MI455X.
// Irregular, bandwidth-bound pipeline: no dense matmul => no WMMA use case.
// CDNA5-relevant paths used instead:
//   * async global->LDS tile staging (GLOBAL_LOAD_ASYNC_TO_LDS_B128 +
//     s_wait_asynccnt), guarded by __has_builtin with a direct-load fallback
//   * wave32 ballot-based stable radix ranking (32-bit EXEC paths)
//   * LDS histograms / per-wave digit counts (ds_* ops)
//   * __builtin_prefetch -> global_prefetch_b8 for streaming tet indices
// Workspace requirement: ~180 MB for Nv=300k, Nt=1.5M.
// ---------------------------------------------------------------------------

typedef unsigned int u32;
typedef unsigned long long u64;

#if defined(__HIP_DEVICE_COMPILE__) &&                                        \
    __has_builtin(__builtin_amdgcn_global_load_async_to_lds_b128) &&          \
    __has_builtin(__builtin_amdgcn_s_wait_asynccnt)
#define USE_ASYNC_LDS 1
#else
#define USE_ASYNC_LDS 0
#endif

#if USE_ASYNC_LDS
// clang's builtin signature (from the round-2 diagnostic):
//   void __builtin_amdgcn_global_load_async_to_lds_b128(
//       __attribute__((address_space(1))) v4i* gsrc,
//       __attribute__((address_space(3))) v4i* ldst, int imm_off, int cpol)
typedef int v4i_t __attribute__((ext_vector_type(4)));
typedef __attribute__((address_space(1))) v4i_t* gas_b128;
typedef __attribute__((address_space(3))) v4i_t* las_b128;
#endif

// Stage one 16B element per lane from global into LDS asynchronously.
__device__ __forceinline__ void async_stage_b128(const void* gsrc, void* ldst) {
#if USE_ASYNC_LDS
    __builtin_amdgcn_global_load_async_to_lds_b128(
        (gas_b128)gsrc, (las_b128)ldst, 0, 0);
#else
    (void)gsrc; (void)ldst;
#endif
}

__device__ __forceinline__ void async_wait() {
#if USE_ASYNC_LDS
    __builtin_amdgcn_s_wait_asynccnt(0);
#endif
}

__device__ __constant__ int d_TRI[16][6] = {
    {-1,-1,-1,-1,-1,-1},{1,0,2,-1,-1,-1},{4,0,3,-1,-1,-1},{1,4,2,1,3,4},
    {3,1,5,-1,-1,-1},{2,3,0,2,5,3},{1,4,0,1,5,4},{4,2,5,-1,-1,-1},
    {4,5,2,-1,-1,-1},{4,1,0,4,5,1},{3,2,0,3,5,2},{1,3,5,-1,-1,-1},
    {4,1,2,4,3,1},{3,0,4,-1,-1,-1},{2,0,1,-1,-1,-1},{-1,-1,-1,-1,-1,-1}};
__device__ __constant__ int d_NTRI[16] = {0,1,1,2,1,2,2,1,1,2,2,1,2,1,1,0};
__device__ __constant__ int d_EA[6] = {0,0,0,1,1,2};
__device__ __constant__ int d_EB[6] = {1,2,3,2,3,3};

// hdr[0] = E (crossing-edge count, atomic), hdr[1] = M (unique), hdr[2] = T1
__global__ void k_init_hdr(u32* hdr) {
    if (threadIdx.x < 4) hdr[threadIdx.x] = 0u;
}

__global__ void k_zero_f32(float* p, int n) {
    int i = blockIdx.x * 256 + threadIdx.x;
    if (i < n) p[i] = 0.0f;
}

// pos = verts + tanh(deform)/grid_res ; occ = sdf > 0
__global__ void k_vertex(const float* __restrict__ verts, const float* __restrict__ sdf,
                         const float* __restrict__ deform, const int* __restrict__ pGridRes,
                         float* __restrict__ pos, u32* __restrict__ occ, int Nv) {
    int i = blockIdx.x * 256 + threadIdx.x;
    if (i >= Nv) return;
    float inv = 1.0f / (float)(*pGridRes);
    #pragma unroll
    for (int c = 0; c < 3; ++c)
        pos[3 * (size_t)i + c] = verts[3 * (size_t)i + c] + tanhf(deform[3 * (size_t)i + c]) * inv;
    occ[i] = (sdf[i] > 0.0f) ? 1u : 0u;
}

// Per-tet: classify (num_tri flags for stable face ordering) and emit crossing
// edge keys (order-free, canonicalized by the sort). Tet indices are staged
// into LDS via async DMA (one b128 per lane).
__global__ void k_tets(const int* __restrict__ indices, const u32* __restrict__ occ,
                       u32* __restrict__ flags1, u32* __restrict__ flags2,
                       u64* __restrict__ keys, u32* __restrict__ pE, int Nt) {
    __shared__ int4 tile[256];
    int t = threadIdx.x;
    int i = blockIdx.x * 256 + t;
    bool act = i < Nt;
#if USE_ASYNC_LDS
    if (act) async_stage_b128(&((const int4*)indices)[i], &tile[t]);
    async_wait();
    __syncthreads();
    int4 v = act ? tile[t] : make_int4(0, 0, 0, 0);
#else
    if (i + 4096 < Nt)  // gfx1250 global_prefetch_b8: stream the tet index array
        __builtin_prefetch(indices + (size_t)(i + 4096) * 4, 0, 0);
    int4 v = act ? ((const int4*)indices)[i] : make_int4(0, 0, 0, 0);
    (void)tile;
#endif
    if (!act) return;
    u32 o0 = occ[v.x], o1 = occ[v.y], o2 = occ[v.z], o3 = occ[v.w];
    u32 tet = o0 | (o1 << 1) | (o2 << 2) | (o3 << 3);
    int nt = d_NTRI[tet];
    flags1[i] = (nt == 1) ? 1u : 0u;
    flags2[i] = (nt == 2) ? 1u : 0u;
    if (tet == 0u || tet == 15u) return;
    int vv[4] = {v.x, v.y, v.z, v.w};
    u32 oo[4] = {o0, o1, o2, o3};
    #pragma unroll
    for (int e = 0; e < 6; ++e) {
        int a = vv[d_EA[e]], b = vv[d_EB[e]];
        if (oo[d_EA[e]] != oo[d_EB[e]]) {
            int lo = a < b ? a : b, hi = a < b ? b : a;
            u64 key = ((u64)(u32)lo << 20) | (u32)hi;   // 40-bit key (Nv < 2^20)
            u32 p = atomicAdd(pE, 1u);
            keys[p] = key;
        }
    }
}

// ---------------- generic exclusive scan (tile = 256, up to 16.7M elems) ----
__global__ void k_scan_tile(const u32* __restrict__ in, u32* __restrict__ out,
                            u32* __restrict__ sums, int n) {
    __shared__ u32 s[256];
    int t = threadIdx.x;
    int i = blockIdx.x * 256 + t;
    u32 v = (i < n) ? in[i] : 0u;
    s[t] = v; __syncthreads();
    u32 val = v;
    for (int off = 1; off < 256; off <<= 1) {
        u32 x = (t >= off) ? s[t - off] : 0u;
        __syncthreads();
        val += x; s[t] = val;
        __syncthreads();
    }
    if (i < n) out[i] = val - v;                     // exclusive
    if (sums && t == 255) sums[blockIdx.x] = val;    // tile total
}

__global__ void k_scan_add(u32* __restrict__ data, const u32* __restrict__ offs, int n) {
    int i = blockIdx.x * 256 + threadIdx.x;
    if (i < n) data[i] += offs[blockIdx.x];
}

__global__ void k_last_total(const u32* __restrict__ ex, const u32* __restrict__ flg,
                             int n, u32* __restrict__ dst) {
    if (blockIdx.x == 0 && threadIdx.x == 0) *dst = ex[n - 1] + flg[n - 1];
}

// ---------------- LSD radix sort (8-bit digits, 5 passes over 40-bit keys) --
__global__ void k_hist(const u64* __restrict__ src, u32* __restrict__ hist,
                       const u32* __restrict__ pE, int G, int shift) {
    __shared__ u32 bins[256];
    int t = threadIdx.x, blk = blockIdx.x;
    int i = blk * 256 + t;
    bins[t] = 0u; __syncthreads();
    u32 E = *pE;
    if (i < (int)E) {
        u32 d = (u32)(src[i] >> shift) & 255u;
        atomicAdd(&bins[d], 1u);     // ds_add in LDS
    }
    __syncthreads();
    hist[(size_t)t * G + blk] = bins[t];
}

// Stable scatter: per-wave digit match via 8 ballots (wave32), cross-wave
// prefix through LDS per-wave histograms.
__global__ void k_scatter(const u64* __restrict__ src, u64* __restrict__ dst,
                          const u32* __restrict__ histEx, const u32* __restrict__ pE,
                          int G, int shift) {
    __shared__ u32 wh[8][256];
    int t = threadIdx.x, blk = blockIdx.x;
    int i = blk * 256 + t;
    u32 E = *pE;
    bool active = i < (int)E;
    u64 key = active ? src[i] : 0xFFFFFFFFFFFFFFFFull;
    u32 d = (u32)(key >> shift) & 255u;
    #pragma unroll
    for (int w = 0; w < 8; ++w) wh[w][t] = 0u;
    __syncthreads();
    u32 lane = (u32)t & 31u, wv = (u32)t >> 5;
    u32 activeMask = (u32)__ballot(active);
    u32 mask = 0xFFFFFFFFu;
    #pragma unroll
    for (int bit = 0; bit < 8; ++bit) {
        u32 b = (u32)__ballot((int)((d >> bit) & 1u));
        mask &= ((d >> bit) & 1u) ? b : ~b;
    }
    mask &= activeMask;
    u32 lt = (1u << lane) - 1u;
    u32 rankInWave = __popc(mask & lt);
    if (active && lane == (u32)(__ffs((int)mask) - 1))
        wh[wv][d] = (u32)__popc(mask);
    __syncthreads();
    if (active) {
        u32 prefix = 0u;
        for (u32 w = 0; w < wv; ++w) prefix += wh[w][d];
        u32 pos = histEx[(size_t)d * G + blk] + prefix + rankInWave;
        dst[pos] = key;
    }
}

// ---------------- unique ----------------------------------------------------
__global__ void k_uniq_flag(const u64* __restrict__ s, u32* __restrict__ flag,
                            const u32* __restrict__ pE, int Emax) {
    int i = blockIdx.x * 256 + threadIdx.x;
    if (i >= Emax) return;
    u32 E = *pE;
    flag[i] = (i < (int)E) ? ((i == 0 || s[i] != s[i - 1]) ? 1u : 0u) : 0u;
}

__global__ void k_uniq_scatter(const u64* __restrict__ s, const u32* __restrict__ flag,
                               const u32* __restrict__ rank, u64* __restrict__ uniq,
                               const u32* __restrict__ pE, int Emax) {
    int i = blockIdx.x * 256 + threadIdx.x;
    if (i >= Emax) return;
    u32 E = *pE;
    if (i < (int)E && flag[i]) uniq[rank[i]] = s[i];
}

// verts_out[u] = pos[a]*(-sdf[b]) / (sdf[a]-sdf[b]) + pos[b]*sdf[a] / (sdf[a]-sdf[b])
__global__ void k_interp(const u64* __restrict__ uniq, const float* __restrict__ pos,
                         const float* __restrict__ sdf, float* __restrict__ outV,
                         const u32* __restrict__ pM, int Emax) {
    int i = blockIdx.x * 256 + threadIdx.x;
    u32 M = *pM;
    if (i >= (int)M) return;
    u64 key = uniq[i];
    int a = (int)(key >> 20);
    int b = (int)(key & 0xFFFFFu);
    float sa = sdf[a], sb = sdf[b];
    float denom = sa - sb;
    float w0 = -sb / denom, w1 = sa / denom;
    #pragma unroll
    for (int c = 0; c < 3; ++c)
        outV[3 * (size_t)i + c] = pos[3 * (size_t)a + c] * w0 + pos[3 * (size_t)b + c] * w1;
}

// faces: m1 rows (stable order) then m2 rows (2 tris each). Edge id via
// binary search in sorted-unique crossing-edge array. Indices staged via
// async DMA into LDS as in k_tets.
__global__ void k_faces(const int* __restrict__ indices, const u32* __restrict__ occ,
                        const u32* __restrict__ r1, const u32* __restrict__ r2,
                        const u64* __restrict__ uniq, const u32* __restrict__ hdr,
                        float* __restrict__ out, int Nt) {
    __shared__ int4 tile[256];
    int t = threadIdx.x;
    int i = blockIdx.x * 256 + t;
    bool act = i < Nt;
#if USE_ASYNC_LDS
    if (act) async_stage_b128(&((const int4*)indices)[i], &tile[t]);
    async_wait();
    __syncthreads();
    int4 v = act ? tile[t] : make_int4(0, 0, 0, 0);
#else
    int4 v = act ? ((const int4*)indices)[i] : make_int4(0, 0, 0, 0);
    (void)tile;
#endif
    if (!act) return;
    u32 o0 = occ[v.x], o1 = occ[v.y], o2 = occ[v.z], o3 = occ[v.w];
    u32 tet = o0 | (o1 << 1) | (o2 << 2) | (o3 << 3);
    int nt = d_NTRI[tet];
    if (nt == 0) return;
    u32 M = hdr[1], T1 = hdr[2];
    float* faces = out + (size_t)3 * M;
    int vv[4] = {v.x, v.y, v.z, v.w};
    size_t base = (nt == 1) ? (size_t)r1[i] * 3
                            : ((size_t)T1 + (size_t)2 * r2[i]) * 3;
    int nE = nt * 3;
    for (int k = 0; k < nE; ++k) {
        int e = d_TRI[tet][k];
        int a = vv[d_EA[e]], b = vv[d_EB[e]];
        int lo = a < b ? a : b, hi = a < b ? b : a;
        u64 key = ((u64)(u32)lo << 20) | (u32)hi;
        u32 loI = 0, hiI = M;
        while (loI < hiI) {
            u32 mid = (loI + hiI) >> 1;
            if (uniq[mid] < key) loI = mid + 1; else hiI = mid;
        }
        faces[base + k] = (float)loI;
    }
}

// ---------------------------------------------------------------------------
static void scan_excl(hipStream_t st, const u32* in, u32* out, int n,
                      u32* s0, u32* s0x, u32* s1, u32* s1x) {
    int B0 = (n + 255) / 256;
    k_scan_tile<<<B0, 256, 0, st>>>(in, out, s0, n);
    if (B0 > 1) {
        int B1 = (B0 + 255) / 256;
        k_scan_tile<<<B1, 256, 0, st>>>(s0, s0x, s1, B0);
        if (B1 > 1) {
            k_scan_tile<<<1, 256, 0, st>>>(s1, s1x, nullptr, B1);
            k_scan_add<<<B1, 256, 0, st>>>(s0x, s1x, B0);
        }
        k_scan_add<<<B0, 256, 0, st>>>(out, s0x, n);
    }
}

extern "C" void kernel_launch(void* const* d_in, const int* in_sizes, int n_in,
                              void* d_out, int out_size, void* d_ws, size_t ws_size,
                              hipStream_t stream) {
    const float* verts  = (const float*)d_in[0];
    const float* sdf    = (const float*)d_in[1];
    const float* deform = (const float*)d_in[2];
    const int*   idx    = (const int*)d_in[3];
    const int*   gres   = (const int*)d_in[4];
    float* out = (float*)d_out;

    const int Nv = in_sizes[1];
    const int Nt = in_sizes[3] / 4;
    const int Emax = 4 * Nt;                  // <=4 crossing edges per tet
    const int G = (Emax + 255) / 256;
    const int histN = 256 * G;

    // ---- workspace carve-out (256B aligned) ----
    char* wp = (char*)d_ws;
    auto alloc = [&](size_t bytes) -> void* {
        void* r = (void*)wp;
        wp += (bytes + 255) & ~(size_t)255;
        return r;
    };
    u32* hdr    = (u32*)alloc(4 * sizeof(u32));
    float* pos  = (float*)alloc((size_t)Nv * 3 * sizeof(float));
    u32* occ    = (u32*)alloc((size_t)Nv * sizeof(u32));
    u32* flags1 = (u32*)alloc((size_t)Nt * sizeof(u32));
    u32* r1     = (u32*)alloc((size_t)Nt * sizeof(u32));
    u32* flags2 = (u32*)alloc((size_t)Nt * sizeof(u32));
    u32* r2     = (u32*)alloc((size_t)Nt * sizeof(u32));
    u64* keyA   = (u64*)alloc((size_t)Emax * sizeof(u64));
    u64* keyB   = (u64*)alloc((size_t)Emax * sizeof(u64));
    u32* shared = (u32*)alloc((size_t)2 * histN * sizeof(u32)); // hist+histEx / flag+rank
    u32* s0  = (u32*)alloc(32768 * sizeof(u32));
    u32* s0x = (u32*)alloc(32768 * sizeof(u32));
    u32* s1  = (u32*)alloc(4096 * sizeof(u32));
    u32* s1x = (u32*)alloc(4096 * sizeof(u32));
    u32* hist   = shared;
    u32* histEx = shared + histN;
    u32* uFlag  = shared;           // reused after sort completes
    u32* uRank  = shared + histN;

    // 1. reset device counters, zero output
    k_init_hdr<<<1, 32, 0, stream>>>(hdr);
    k_zero_f32<<<(out_size + 255) / 256, 256, 0, stream>>>(out, out_size);

    // 2. per-vertex: deformed pos + occupancy
    k_vertex<<<(Nv + 255) / 256, 256, 0, stream>>>(verts, sdf, deform, gres, pos, occ, Nv);

    // 3. per-tet classify + emit crossing-edge keys
    k_tets<<<(Nt + 255) / 256, 256, 0, stream>>>(idx, occ, flags1, flags2, keyA, &hdr[0], Nt);

    // 4. stable face-row ranks
    scan_excl(stream, flags1, r1, Nt, s0, s0x, s1, s1x);
    k_last_total<<<1, 32, 0, stream>>>(r1, flags1, Nt, &hdr[2]);   // T1
    scan_excl(stream, flags2, r2, Nt, s0, s0x, s1, s1x);

    // 5. LSD radix sort: 5 passes over 40-bit keys, A<->B ping-pong
    u64* src = keyA;
    u64* dst = keyB;
    for (int p = 0; p < 5; ++p) {
        int shift = 8 * p;
        k_hist<<<G, 256, 0, stream>>>(src, hist, &hdr[0], G, shift);
        scan_excl(stream, hist, histEx, histN, s0, s0x, s1, s1x);
        k_scatter<<<G, 256, 0, stream>>>(src, dst, histEx, &hdr[0], G, shift);
        u64* t = src; src = dst; dst = t;
    }
    // sorted data now in `src` (== keyB after 5 passes); keyA is free

    // 6. unique (crossing-edge ids in sorted order == reference mapping)
    k_uniq_flag<<<(Emax + 255) / 256, 256, 0, stream>>>(src, uFlag, &hdr[0], Emax);
    scan_excl(stream, uFlag, uRank, Emax, s0, s0x, s1, s1x);
    k_last_total<<<1, 32, 0, stream>>>(uRank, uFlag, Emax, &hdr[1]); // M
    k_uniq_scatter<<<(Emax + 255) / 256, 256, 0, stream>>>(src, uFlag, uRank, keyA, &hdr[0], Emax);

    // 7. interpolated surface vertices
    k_interp<<<(Emax + 255) / 256, 256, 0, stream>>>(keyA, pos, sdf, out, &hdr[1], Emax);

    // 8. faces via triangle table + binary search into unique edges
    k_faces<<<(Nt + 255) / 256, 256, 0, stream>>>(idx, occ, r1, r2, keyA, hdr, out, Nt);
}